// Attention_84653805404827
// MI455X (gfx1250) — compile-verified
//
#include <hip/hip_runtime.h>
#include <hip/hip_bf16.h>
#include <math.h>

// Problem constants (match reference)
#define B_   4
#define SQ_  2048
#define SKV_ 2048
#define DM_  1024
#define DK_  64
#define NH_  8
#define KC_  64      // K-chunk staged through LDS (64 x 64 f32 = 16 KB)

typedef __attribute__((ext_vector_type(2))) float v2f;
typedef __attribute__((ext_vector_type(8))) float v8f;

// D = A(16x4 f32) * B(4x16 f32) + C(16x16 f32)
__device__ __forceinline__ v8f wmma4(v2f a, v2f b, v8f c) {
  return __builtin_amdgcn_wmma_f32_16x16x4_f32(
      /*neg_a=*/false, a, /*neg_b=*/false, b,
      /*c_mod=*/(short)0, c, /*reuse_a=*/false, /*reuse_b=*/false);
}

// 16-byte async copy global -> LDS (tracked by ASYNCcnt). lds_addr is the LDS
// byte offset (low 32 bits of the generic pointer per the LDS aperture rule).
__device__ __forceinline__ void async_copy_b128(unsigned lds_addr, const void* gptr) {
  asm volatile("global_load_async_to_lds_b128 %0, %1, off"
               :: "v"(lds_addr), "v"((unsigned long long)(uintptr_t)gptr)
               : "memory");
}
__device__ __forceinline__ void wait_async0() {
  asm volatile("s_wait_asynccnt 0" ::: "memory");
}
__device__ __forceinline__ unsigned lds_off(const void* p) {
  return (unsigned)(uintptr_t)p;     // LDS generic addr truncates to byte offset
}

// Issue one 64x64 f32 chunk copy (16 KB): 4 x B128 per thread, 256 threads.
__device__ __forceinline__ void stage_chunk(unsigned lds_base, const float* gsrc,
                                            unsigned tid) {
  #pragma unroll
  for (int j = 0; j < 4; ++j) {
    const unsigned foff = (unsigned)j * 1024u + tid * 4u;     // float index
    async_copy_b128(lds_base + foff * 4u, (const void*)(gsrc + foff));
  }
}

// dst[M,64] = src[M,1024] @ W[1024,64]; one wave per 16-row tile, 8 waves/block.
// W streamed through a double-buffered LDS ping-pong (async copy overlaps WMMA).
__global__ __launch_bounds__(256) void proj_kernel(const float* __restrict__ src,
                                                   const float* __restrict__ W,
                                                   float* __restrict__ dst) {
  __shared__ float ldsW[2][KC_ * DK_];               // 32 KB
  const unsigned tid  = threadIdx.x;
  const unsigned lane = tid & 31u;
  const unsigned wid  = tid >> 5;
  const unsigned l16  = lane & 15u;
  const unsigned half = lane >> 4;                   // 0 or 1
  const unsigned rowbase = (blockIdx.x * 8u + wid) * 16u;
  const unsigned ldsb[2] = { lds_off(ldsW[0]), lds_off(ldsW[1]) };

  v8f acc[4] = {};
  const float* arow = src + (size_t)(rowbase + l16) * DM_ + 2u * half;

  const int NC = DM_ / KC_;                          // 16 chunks
  stage_chunk(ldsb[0], W, tid);                      // prologue: chunk 0
  for (int c = 0; c < NC; ++c) {
    const unsigned cur = (unsigned)c & 1u;
    wait_async0();
    __syncthreads();                                 // buf[cur] visible to block
    if (c + 1 < NC)                                  // overlap next fill w/ math
      stage_chunk(ldsb[cur ^ 1u], W + (size_t)(c + 1) * KC_ * DK_, tid);

    const int kc = c * KC_;
    for (int k = 0; k < KC_; k += 4) {
      v2f a = *(const v2f*)(arow + kc + k);
      const float* wrow = ldsW[cur] + (size_t)(k + 2u * half) * DK_ + l16;
      #pragma unroll
      for (int n = 0; n < 4; ++n) {
        v2f b;
        b.x = wrow[n * 16];
        b.y = wrow[n * 16 + DK_];
        acc[n] = wmma4(a, b, acc[n]);
      }
    }
    __syncthreads();                                 // block done reading buf[cur]
  }
  #pragma unroll
  for (int n = 0; n < 4; ++n) {
    #pragma unroll
    for (int i = 0; i < 8; ++i) {
      const unsigned row = rowbase + (unsigned)i + 8u * half;
      dst[(size_t)row * DK_ + n * 16 + l16] = acc[n][i];
    }
  }
}

// raw[b,q,k] = w8*(masked scale*Q.K^T) + sum_h w[h]*prev[h,b,q,k], NaN->-inf.
// The block's shared 16-row Q tile is staged through LDS (async B128).
__global__ __launch_bounds__(256) void scores_kernel(
    const float* __restrict__ Q, const float* __restrict__ K,
    const float* __restrict__ prev,
    const unsigned char* __restrict__ qmask, const unsigned char* __restrict__ kmask,
    const float* __restrict__ resw,
    float* __restrict__ raw) {
  __shared__ float ldsQ[16 * DK_];                   // 4 KB
  const unsigned tid  = threadIdx.x;
  const unsigned lane = tid & 31u;
  const unsigned wid  = tid >> 5;
  const unsigned l16  = lane & 15u;
  const unsigned half = lane >> 4;
  const unsigned kbase = (blockIdx.x * 8u + wid) * 16u;
  const unsigned qbase = blockIdx.y * 16u;
  const unsigned bb    = blockIdx.z;

  // Stage Q tile: 16 rows x 64 f32 = 4096 B = 256 threads x 16 B.
  async_copy_b128(lds_off(ldsQ) + tid * 16u,
                  (const void*)(Q + ((size_t)bb * SQ_ + qbase) * DK_ + tid * 4u));

  float w[NH_ + 1];
  float wsum = 0.f;
  #pragma unroll
  for (int h = 0; h <= NH_; ++h) { w[h] = resw[h]; wsum += w[h]; }
  const float winv = 1.0f / wsum;
  #pragma unroll
  for (int h = 0; h <= NH_; ++h) w[h] *= winv;

  wait_async0();
  __syncthreads();

  v8f acc = {};
  const float* qp = ldsQ + (size_t)l16 * DK_ + 2u * half;
  const float* kp = K + ((size_t)bb * SKV_ + kbase + l16) * DK_ + 2u * half;
  #pragma unroll
  for (int d = 0; d < DK_; d += 4) {
    v2f a = *(const v2f*)(qp + d);   // A row = Q row (from LDS)
    v2f b = *(const v2f*)(kp + d);   // B col n = K row (K^T), depth contiguous
    acc = wmma4(a, b, acc);
  }

  const unsigned col = kbase + l16;
  const bool kmv = kmask[(size_t)bb * SKV_ + col] != 0;
  const float NEGINF = -__builtin_inff();
  #pragma unroll
  for (int i = 0; i < 8; ++i) {
    const unsigned row = qbase + (unsigned)i + 8u * half;
    float s = acc[i] * 0.125f;                        // D_K^-0.5
    if (kmv || (qmask[(size_t)bb * SQ_ + row] != 0)) s = NEGINF;
    float r = w[NH_] * s;
    #pragma unroll
    for (int h = 0; h < NH_; ++h) {
      if (w[h] != 0.0f)                               // wave-uniform skip
        r += w[h] * prev[(((size_t)h * B_ + bb) * SQ_ + row) * SKV_ + col];
    }
    if (r != r) r = NEGINF;                           // NaN -> -inf
    raw[((size_t)bb * SQ_ + row) * SKV_ + col] = r;
  }
}

// Per-row softmax stats: guarded max and reciprocal sum-of-exp (wave32 shuffles).
__global__ __launch_bounds__(256) void stats_kernel(const float* __restrict__ raw,
                                                    float* __restrict__ rowmax,
                                                    float* __restrict__ rowscale) {
  const unsigned lane = threadIdx.x & 31u;
  const unsigned wid  = threadIdx.x >> 5;
  const unsigned row  = blockIdx.x * 8u + wid;
  const unsigned bb   = blockIdx.y;
  const float* rp = raw + ((size_t)bb * SQ_ + row) * SKV_;

  float m = -__builtin_inff();
  for (int c = (int)lane; c < SKV_; c += 32) m = fmaxf(m, rp[c]);
  #pragma unroll
  for (int off = 16; off >= 1; off >>= 1) m = fmaxf(m, __shfl_xor(m, off, 32));
  const float mr = (m > -__builtin_inff()) ? m : 0.0f; // fully-masked row guard

  float s = 0.f;
  for (int c = (int)lane; c < SKV_; c += 32) s += __expf(rp[c] - mr);
  #pragma unroll
  for (int off = 16; off >= 1; off >>= 1) s += __shfl_xor(s, off, 32);

  if (lane == 0) {
    rowmax[(size_t)bb * SQ_ + row]   = mr;
    rowscale[(size_t)bb * SQ_ + row] = (s > 0.f) ? (1.0f / s) : 0.0f;
  }
}

// out[b,q,:] = softmax(raw[b,q,:]) @ V. exp() applied while building the A
// fragment; 1/sum applied at store. V streamed via double-buffered LDS ping-pong.
__global__ __launch_bounds__(256) void av_kernel(const float* __restrict__ raw,
                                                 const float* __restrict__ V,
                                                 const float* __restrict__ rowmax,
                                                 const float* __restrict__ rowscale,
                                                 float* __restrict__ out) {
  __shared__ float ldsV[2][KC_ * DK_];               // 32 KB
  const unsigned tid  = threadIdx.x;
  const unsigned lane = tid & 31u;
  const unsigned wid  = tid >> 5;
  const unsigned l16  = lane & 15u;
  const unsigned half = lane >> 4;
  const unsigned qbase = (blockIdx.x * 8u + wid) * 16u;
  const unsigned bb    = blockIdx.y;
  const unsigned ldsb[2] = { lds_off(ldsV[0]), lds_off(ldsV[1]) };

  const float rm = rowmax[(size_t)bb * SQ_ + qbase + l16];   // A-fragment row
  const float* rp = raw + ((size_t)bb * SQ_ + qbase + l16) * SKV_ + 2u * half;
  const float* vb = V + (size_t)bb * SKV_ * DK_;

  v8f acc[4] = {};
  const int NC = SKV_ / KC_;                         // 32 chunks
  stage_chunk(ldsb[0], vb, tid);                     // prologue: chunk 0
  for (int c = 0; c < NC; ++c) {
    const unsigned cur = (unsigned)c & 1u;
    wait_async0();
    __syncthreads();                                 // buf[cur] visible to block
    if (c + 1 < NC)                                  // overlap next fill w/ math
      stage_chunk(ldsb[cur ^ 1u], vb + (size_t)(c + 1) * KC_ * DK_, tid);

    const int kc = c * KC_;
    for (int k = 0; k < KC_; k += 4) {
      v2f rr = *(const v2f*)(rp + kc + k);
      v2f a;
      a.x = __expf(rr.x - rm);   // fully-masked: raw=-inf, rm=0 -> exp(-inf)=0
      a.y = __expf(rr.y - rm);
      const float* vrow = ldsV[cur] + (size_t)(k + 2u * half) * DK_ + l16;
      #pragma unroll
      for (int n = 0; n < 4; ++n) {
        v2f b;
        b.x = vrow[n * 16];
        b.y = vrow[n * 16 + DK_];
        acc[n] = wmma4(a, b, acc[n]);
      }
    }
    __syncthreads();                                 // block done reading buf[cur]
  }
  #pragma unroll
  for (int i = 0; i < 8; ++i) {
    const unsigned row = qbase + (unsigned)i + 8u * half;
    const float sc = rowscale[(size_t)bb * SQ_ + row];
    #pragma unroll
    for (int n = 0; n < 4; ++n)
      out[((size_t)bb * SQ_ + row) * DK_ + n * 16 + l16] = acc[n][i] * sc;
  }
}

extern "C" void kernel_launch(void* const* d_in, const int* in_sizes, int n_in,
                              void* d_out, int out_size, void* d_ws, size_t ws_size,
                              hipStream_t stream) {
  (void)in_sizes; (void)n_in; (void)out_size; (void)ws_size;
  const float*         src_q  = (const float*)d_in[0];
  const float*         src_kv = (const float*)d_in[1];
  const unsigned char* qmask  = (const unsigned char*)d_in[2];
  const unsigned char* kmask  = (const unsigned char*)d_in[3];
  const float*         prev   = (const float*)d_in[4];
  const float*         Wq     = (const float*)d_in[5];
  const float*         Wk     = (const float*)d_in[6];
  const float*         Wv     = (const float*)d_in[7];
  const float*         resw   = (const float*)d_in[8];

  float* outp = (float*)d_out;                        // [B,SQ,DV]
  float* rawp = outp + (size_t)B_ * SQ_ * DK_;        // [B,SQ,SKV]

  float* ws     = (float*)d_ws;                       // ~6.1 MB used
  float* Qp     = ws;
  float* Kp     = Qp + (size_t)B_ * SQ_  * DK_;
  float* Vp     = Kp + (size_t)B_ * SKV_ * DK_;
  float* rmax   = Vp + (size_t)B_ * SKV_ * DK_;
  float* rscale = rmax + (size_t)B_ * SQ_;

  dim3 blk(256);
  const int MROWS = B_ * SQ_;                         // 8192
  proj_kernel<<<dim3(MROWS / 128), blk, 0, stream>>>(src_q,  Wq, Qp);
  proj_kernel<<<dim3(MROWS / 128), blk, 0, stream>>>(src_kv, Wk, Kp);
  proj_kernel<<<dim3(MROWS / 128), blk, 0, stream>>>(src_kv, Wv, Vp);
  scores_kernel<<<dim3(SKV_ / 128, SQ_ / 16, B_), blk, 0, stream>>>(
      Qp, Kp, prev, qmask, kmask, resw, rawp);
  stats_kernel<<<dim3(SQ_ / 8, B_), blk, 0, stream>>>(rawp, rmax, rscale);
  av_kernel<<<dim3(SQ_ / 128, B_), blk, 0, stream>>>(rawp, Vp, rmax, rscale, outp);
}